// HNMDiscriminativeLoss_32229434589497
// MI455X (gfx1250) — compile-verified
//
#include <hip/hip_runtime.h>
#include <hip/hip_bf16.h>

// ---------------------------------------------------------------------------
// HNM discriminative loss for MI455X (gfx1250, wave32).
// predict: (4,32,512,1024) f32, target: (4,512,1024) i32, out: scalar f32.
// Memory-bound: 2 streaming passes over predict (~528MB => ~23us @ 23.3TB/s).
// lane==channel mapping makes seg wave-uniform => scalar-branch accumulation.
// Centers Gram matrix computed with V_WMMA_F32_16X16X4_F32.
// ---------------------------------------------------------------------------

#define KCLS 19
#define CCH  32
#define HW   (512 * 1024)      // 2^19
#define HWSH 19
#define NPIX (4 * HW)          // 2,097,152
#define NBLK 2048              // streaming-pass blocks (256 thr = 8 waves each)
#define PIX_PER_WAVE (NPIX / (NBLK * 8))   // 128
#define QUADS (PIX_PER_WAVE / 4)           // 32

// workspace offsets (floats)
#define WS_SUMS    0      // 19*32 = 608
#define WS_COUNTS  640    // 19 (as unsigned)
#define WS_CENTERS 704    // 19*32 = 608
#define WS_SQ      1344   // 19
#define WS_POS     1376   // 19
#define WS_VALID   1408   // 19
#define WS_MISC    1440   // [0]=loss_dis, [1]=0.001*loss_reg/n_cls, [2]=n_cls
#define WS_ZERO_N  1536

typedef __attribute__((ext_vector_type(2))) float v2f;
typedef __attribute__((ext_vector_type(8))) float v8f;

#define CASES19(X) X(0) X(1) X(2) X(3) X(4) X(5) X(6) X(7) X(8) X(9) \
                   X(10) X(11) X(12) X(13) X(14) X(15) X(16) X(17) X(18)

// ---------------------------------------------------------------- zero ws ---
__global__ void k_zero(float* ws) {
    for (int t = threadIdx.x; t < WS_ZERO_N; t += blockDim.x) ws[t] = 0.0f;
}

// ------------------------------------------------------- class histogram ----
__global__ void k_hist(const int* __restrict__ tgt, unsigned* __restrict__ cnt, int n) {
    __shared__ unsigned h[32];
    int tid = threadIdx.x;
    if (tid < 32) h[tid] = 0u;
    __syncthreads();
    for (int i = blockIdx.x * blockDim.x + tid; i < n; i += gridDim.x * blockDim.x) {
        int s = tgt[i];
        if ((unsigned)s < (unsigned)KCLS) atomicAdd(&h[s], 1u);
    }
    __syncthreads();
    if (tid < KCLS) atomicAdd(&cnt[tid], h[tid]);
}

// -------------------------------------------- pass 1: per-class sums --------
__global__ __launch_bounds__(256) void k_sums(const float* __restrict__ pred,
                                              const int* __restrict__ tgt,
                                              float* __restrict__ gsums) {
    __shared__ float ls[KCLS * CCH];
    const int tid = threadIdx.x;
    for (int t = tid; t < KCLS * CCH; t += 256) ls[t] = 0.0f;
    __syncthreads();

    const int lane = tid & 31;                       // lane == channel
    const int wave = blockIdx.x * 8 + (tid >> 5);
    const long p0  = (long)wave * PIX_PER_WAVE;      // contiguous pixel chunk
    const int  n   = (int)(p0 >> HWSH);
    const int  hw0 = (int)(p0 & (HW - 1));
    const float* pp = pred + (((long)(n * CCH + lane)) << HWSH) + hw0;
    const int*   sp = tgt + p0;

    float acc[KCLS];
#pragma unroll
    for (int k = 0; k < KCLS; ++k) acc[k] = 0.0f;

#define ACC_CASE(k) case k: acc[k] += v_; break;
#define DO_ACC(s_, vv) { float v_ = (vv); switch (s_) { CASES19(ACC_CASE) default: break; } }

    for (int q = 0; q < QUADS; ++q) {
        const float4 f  = *(const float4*)(pp + q * 4);
        const int4   s4 = *(const int4*)(sp + q * 4);
        __builtin_prefetch(pp + q * 4 + 64, 0, 0);   // global_prefetch_b8, 256B ahead
        int s;
        s = __builtin_amdgcn_readfirstlane(s4.x); DO_ACC(s, f.x);
        s = __builtin_amdgcn_readfirstlane(s4.y); DO_ACC(s, f.y);
        s = __builtin_amdgcn_readfirstlane(s4.z); DO_ACC(s, f.z);
        s = __builtin_amdgcn_readfirstlane(s4.w); DO_ACC(s, f.w);
    }
#undef DO_ACC
#undef ACC_CASE

#pragma unroll
    for (int k = 0; k < KCLS; ++k) atomicAdd(&ls[k * CCH + lane], acc[k]);
    __syncthreads();
    for (int t = tid; t < KCLS * CCH; t += 256) atomicAdd(&gsums[t], ls[t]);
}

// -------------------- centers + WMMA Gram + pairwise/reg terms (1 wave) -----
__global__ __launch_bounds__(32) void k_centers(const float* __restrict__ gsums,
                                                const unsigned* __restrict__ gcnt,
                                                float* __restrict__ gctr,
                                                float* __restrict__ validf,
                                                float* __restrict__ misc) {
    __shared__ float Cm[32 * 32];   // padded centers (rows 19..31 zero)
    __shared__ float Gm[32 * 32];   // Gram = Cm * Cm^T
    __shared__ float CNT[32];
    __shared__ float VAL[32];
    const int lane = threadIdx.x;

    float cnt = (lane < KCLS) ? (float)gcnt[lane] : 0.0f;
    CNT[lane] = cnt;
    float val = (lane < KCLS && cnt > 20.0f) ? 1.0f : 0.0f;
    VAL[lane] = val;
    float nv = val;
#pragma unroll
    for (int off = 16; off >= 1; off >>= 1) nv += __shfl_xor(nv, off, 32);
    const float n_cls = fmaxf(nv, 1.0f);
    __syncthreads();

    for (int k = 0; k < 32; ++k) {                   // lane == channel
        float c = 0.0f;
        if (k < KCLS) c = gsums[k * 32 + lane] / fmaxf(CNT[k], 1.0f);
        Cm[k * 32 + lane] = c;
        if (k < KCLS) gctr[k * 32 + lane] = c;
    }
    if (lane < KCLS) validf[lane] = val;
    __syncthreads();

    // Gram via V_WMMA_F32_16X16X4_F32: D[16x16] += A[16x4] * B[4x16]
    // A: lane l -> row m = l&15; vgpr j -> k = 2*(l>>4) + j. B symmetric (B = C^T).
    const int half = lane >> 4, mr = lane & 15;
    for (int mt = 0; mt < 32; mt += 16)
    for (int nt = 0; nt < 32; nt += 16) {
        v8f acc = {};
#pragma unroll
        for (int kb = 0; kb < 32; kb += 4) {
            const int k0 = kb + 2 * half;
            v2f a, b;
            a.x = Cm[(mt + mr) * 32 + k0];
            a.y = Cm[(mt + mr) * 32 + k0 + 1];
            b.x = Cm[(nt + mr) * 32 + k0];
            b.y = Cm[(nt + mr) * 32 + k0 + 1];
            acc = __builtin_amdgcn_wmma_f32_16x16x4_f32(
                false, a, false, b, (short)0, acc, false, false);
        }
#pragma unroll
        for (int r = 0; r < 8; ++r)
            Gm[(mt + r + 8 * half) * 32 + (nt + mr)] = acc[r];
    }
    __syncthreads();

    // pairwise: relu(2*DELTA - dist)^2 over valid i!=j
    float dsum = 0.0f;
    for (int p = lane; p < KCLS * KCLS; p += 32) {
        const int i = p / KCLS, j = p % KCLS;
        if (i != j && VAL[i] > 0.5f && VAL[j] > 0.5f) {
            float d2 = Gm[i * 32 + i] + Gm[j * 32 + j] - 2.0f * Gm[i * 32 + j];
            float dist = sqrtf(fmaxf(d2, 0.0f) + 1e-12f);
            float t = 3.0f - dist;                   // 2*DELTA = 3.0
            if (t > 0.0f) dsum += t * t;
        }
    }
#pragma unroll
    for (int off = 16; off >= 1; off >>= 1) dsum += __shfl_xor(dsum, off, 32);

    float rsum = (lane < KCLS && val > 0.5f) ? sqrtf(Gm[lane * 32 + lane] + 1e-12f) : 0.0f;
#pragma unroll
    for (int off = 16; off >= 1; off >>= 1) rsum += __shfl_xor(rsum, off, 32);

    if (lane == 0) {
        misc[0] = dsum / fmaxf(n_cls * (n_cls - 1.0f), 1.0f);
        misc[1] = 0.001f * rsum / n_cls;
        misc[2] = n_cls;
    }
}

// ------------------- pass 2: variance term (sq & pos per class) -------------
__global__ __launch_bounds__(256) void k_var(const float* __restrict__ pred,
                                             const int* __restrict__ tgt,
                                             const float* __restrict__ gctr,
                                             float* __restrict__ gsq,
                                             float* __restrict__ gpos) {
    __shared__ float ls[2 * KCLS];
    const int tid = threadIdx.x;
    for (int t = tid; t < 2 * KCLS; t += 256) ls[t] = 0.0f;
    __syncthreads();

    const int lane = tid & 31;                       // lane == channel
    const int wave = blockIdx.x * 8 + (tid >> 5);
    const long p0  = (long)wave * PIX_PER_WAVE;
    const int  n   = (int)(p0 >> HWSH);
    const int  hw0 = (int)(p0 & (HW - 1));
    const float* pp = pred + (((long)(n * CCH + lane)) << HWSH) + hw0;
    const int*   sp = tgt + p0;

    float ctr[KCLS];
#pragma unroll
    for (int k = 0; k < KCLS; ++k) ctr[k] = gctr[k * CCH + lane];

    float sq[KCLS], ps[KCLS];
#pragma unroll
    for (int k = 0; k < KCLS; ++k) { sq[k] = 0.0f; ps[k] = 0.0f; }

#define SEL_CASE(k) case k: cv_ = ctr[k]; break;
#define DO_SEL(s_, dv, fv) { float cv_ = (fv); switch (s_) { CASES19(SEL_CASE) default: break; } dv = cv_ - (fv); }
#define ACC2_CASE(k) case k: sq[k] += r2_; ps[k] += 1.0f; break;
#define DO_ACC2(s_, r2v) { float r2_ = (r2v); switch (s_) { CASES19(ACC2_CASE) default: break; } }

    for (int q = 0; q < QUADS; ++q) {
        const float4 f  = *(const float4*)(pp + q * 4);
        const int4   s4 = *(const int4*)(sp + q * 4);
        __builtin_prefetch(pp + q * 4 + 64, 0, 0);
        const int s0 = __builtin_amdgcn_readfirstlane(s4.x);
        const int s1 = __builtin_amdgcn_readfirstlane(s4.y);
        const int s2 = __builtin_amdgcn_readfirstlane(s4.z);
        const int s3 = __builtin_amdgcn_readfirstlane(s4.w);
        float d0, d1, d2, d3;
        DO_SEL(s0, d0, f.x);   // invalid class -> cv_==fv -> d==0 -> r==0
        DO_SEL(s1, d1, f.y);
        DO_SEL(s2, d2, f.z);
        DO_SEL(s3, d3, f.w);
        float t0 = d0 * d0, t1 = d1 * d1, t2 = d2 * d2, t3 = d3 * d3;
#pragma unroll
        for (int off = 16; off >= 1; off >>= 1) {    // cross-channel butterfly
            t0 += __shfl_xor(t0, off, 32);
            t1 += __shfl_xor(t1, off, 32);
            t2 += __shfl_xor(t2, off, 32);
            t3 += __shfl_xor(t3, off, 32);
        }
        float r;
        r = sqrtf(t0 + 1e-12f) - 0.5f; if (r > 0.0f) DO_ACC2(s0, r * r);
        r = sqrtf(t1 + 1e-12f) - 0.5f; if (r > 0.0f) DO_ACC2(s1, r * r);
        r = sqrtf(t2 + 1e-12f) - 0.5f; if (r > 0.0f) DO_ACC2(s2, r * r);
        r = sqrtf(t3 + 1e-12f) - 0.5f; if (r > 0.0f) DO_ACC2(s3, r * r);
    }
#undef DO_SEL
#undef SEL_CASE
#undef DO_ACC2
#undef ACC2_CASE

    if (lane == 0) {                                 // values uniform per wave
#pragma unroll
        for (int k = 0; k < KCLS; ++k) {
            atomicAdd(&ls[k], sq[k]);
            atomicAdd(&ls[KCLS + k], ps[k]);
        }
    }
    __syncthreads();
    for (int t = tid; t < KCLS; t += 256) {
        atomicAdd(&gsq[t], ls[t]);
        atomicAdd(&gpos[t], ls[KCLS + t]);
    }
}

// ------------------------------------------------------------- finalize -----
__global__ void k_final(const float* __restrict__ gsq, const float* __restrict__ gpos,
                        const float* __restrict__ validf, const float* __restrict__ misc,
                        float* __restrict__ out) {
    if (threadIdx.x == 0 && blockIdx.x == 0) {
        const float n_cls = misc[2];
        float lv = 0.0f;
        for (int k = 0; k < KCLS; ++k)
            if (validf[k] > 0.5f) lv += gsq[k] / fmaxf(gpos[k], 1.0f);
        lv /= n_cls;
        out[0] = 1.0f * (lv + misc[0] + misc[1]);    // LOSS_WEIGHT = 1
    }
}

// ---------------------------------------------------------------------------
extern "C" void kernel_launch(void* const* d_in, const int* in_sizes, int n_in,
                              void* d_out, int out_size, void* d_ws, size_t ws_size,
                              hipStream_t stream) {
    const float* pred = (const float*)d_in[0];
    const int*   tgt  = (const int*)d_in[1];
    float* ws  = (float*)d_ws;
    float* out = (float*)d_out;

    float*    sums   = ws + WS_SUMS;
    unsigned* counts = (unsigned*)(ws + WS_COUNTS);
    float*    ctrs   = ws + WS_CENTERS;
    float*    sq     = ws + WS_SQ;
    float*    pos    = ws + WS_POS;
    float*    valid  = ws + WS_VALID;
    float*    misc   = ws + WS_MISC;

    k_zero<<<1, 256, 0, stream>>>(ws);
    k_hist<<<512, 256, 0, stream>>>(tgt, counts, NPIX);
    k_sums<<<NBLK, 256, 0, stream>>>(pred, tgt, sums);
    k_centers<<<1, 32, 0, stream>>>(sums, counts, ctrs, valid, misc);
    k_var<<<NBLK, 256, 0, stream>>>(pred, tgt, ctrs, sq, pos);
    k_final<<<1, 32, 0, stream>>>(sq, pos, valid, misc, out);
}